// LinearQuantileRegression_80522046865738
// MI455X (gfx1250) — compile-verified
//
#include <hip/hip_runtime.h>
#include <hip/hip_bf16.h>
#include <stdint.h>

// Nearest-quantile lookup: for each sample b, argmin_g ||Y_surf[b,:,g]-y[b,:]||^2,
// then u[b,:] = U_grid[:, idx]. Memory-bound (0.75 FLOP/byte at 23.3 TB/s):
// 4-deep async global->LDS pipeline (CDNA5 ASYNCcnt path), one wave32 per
// sample, non-temporal loads (Y_surf = 328 MB streamed once > 192 MB L2).

#define WAVE 32
#define WAVES_PER_BLOCK 8
#define BLOCK (WAVE * WAVES_PER_BLOCK)
#define CHUNK 128           // g-values per chunk per wave = 32 lanes * 4 floats
#define BUFS 4              // circular buffer depth (prefetch distance 3)
#define CHUNK_BYTES (CHUNK * 4)            // 512 B per row per chunk
#define WAVE_LDS_BYTES (BUFS * 2 * CHUNK_BYTES)  // 4 KB per wave

__device__ __forceinline__ void async_copy_b128_nt(uint32_t lds_addr, const void* gaddr) {
    // GLOBAL_LOAD_ASYNC_TO_LDS_B128 (GV mode): per-lane 16B global -> LDS,
    // tracked by ASYNCcnt. Non-temporal: stream-once data, don't pollute L2.
    asm volatile("global_load_async_to_lds_b128 %0, %1, off th:TH_LOAD_NT"
                 :: "v"(lds_addr), "v"((unsigned long long)(uintptr_t)gaddr)
                 : "memory");
}
__device__ __forceinline__ void wait_async_6() { asm volatile("s_wait_asynccnt 6" ::: "memory"); }
__device__ __forceinline__ void wait_async_4() { asm volatile("s_wait_asynccnt 4" ::: "memory"); }
__device__ __forceinline__ void wait_async_2() { asm volatile("s_wait_asynccnt 2" ::: "memory"); }
__device__ __forceinline__ void wait_async_0() { asm volatile("s_wait_asynccnt 0" ::: "memory"); }

__global__ __launch_bounds__(BLOCK)
void vq_push_argmin_kernel(const float* __restrict__ y,      // [B,2]
                           const float* __restrict__ Ysurf,  // [B,2,G]
                           const float* __restrict__ Ugrid,  // [2,G]
                           float* __restrict__ out,          // [B,2]
                           int Bn, int G)
{
    // [wave][buffer][row][elem] : 8 * 4 * 2 * 128 floats = 32 KB
    __shared__ float lds[WAVES_PER_BLOCK][BUFS][2][CHUNK];

    const int lane = threadIdx.x & (WAVE - 1);
    const int wid  = threadIdx.x >> 5;
    int b = blockIdx.x * WAVES_PER_BLOCK + wid;
    b = __builtin_amdgcn_readfirstlane(b);   // wave-uniform -> scalar regs
    if (b >= Bn) return;

    const float* row0 = Ysurf + (size_t)b * 2u * (size_t)G;
    const float* row1 = row0 + G;
    const float y0 = y[2 * b];
    const float y1 = y[2 * b + 1];

    const int nchunk = (G + CHUNK - 1) / CHUNK;

    // Wave-relative LDS byte offset: low 32 bits of the flat (generic) address.
    const uint32_t lds_wave_base =
        (uint32_t)(uintptr_t)(&lds[0][0][0][0]) + (uint32_t)wid * WAVE_LDS_BYTES;
    const uint32_t lane_byte = (uint32_t)lane * 16u;

    // Issue one chunk's async loads (2 x b128 per lane). Addresses are clamped
    // so tail lanes stay in-bounds; EXEC stays uniform for the async issue.
    auto issue = [&](int c) {
        int gbase  = c * CHUNK + lane * 4;
        int gclamp = gbase > (G - 4) ? (G - 4) : gbase;
        uint32_t l0 = lds_wave_base + (uint32_t)(c & (BUFS - 1)) * (2u * CHUNK_BYTES) + lane_byte;
        async_copy_b128_nt(l0,               row0 + gclamp);
        async_copy_b128_nt(l0 + CHUNK_BYTES, row1 + gclamp);
    };

    float bestd = __builtin_inff();
    int   bestg = 0x7fffffff;

    auto compute = [&](int c) {
        const int buf   = c & (BUFS - 1);
        const int gbase = c * CHUNK + lane * 4;
        if (gbase <= G - 4) {    // G % 4 == 0: lanes are all-or-nothing valid
            const float4 a0 = *(const float4*)&lds[wid][buf][0][lane * 4];
            const float4 a1 = *(const float4*)&lds[wid][buf][1][lane * 4];
            float dx, dz, d;
            dx = a0.x - y0; dz = a1.x - y1; d = dx * dx + dz * dz;
            if (d < bestd) { bestd = d; bestg = gbase; }
            dx = a0.y - y0; dz = a1.y - y1; d = dx * dx + dz * dz;
            if (d < bestd) { bestd = d; bestg = gbase + 1; }
            dx = a0.z - y0; dz = a1.z - y1; d = dx * dx + dz * dz;
            if (d < bestd) { bestd = d; bestg = gbase + 2; }
            dx = a0.w - y0; dz = a1.w - y1; d = dx * dx + dz * dz;
            if (d < bestd) { bestd = d; bestg = gbase + 3; }
        }
    };

    // Prologue: fill the pipeline (3 chunks ahead).
    for (int i = 0; i < BUFS - 1 && i < nchunk; ++i) issue(i);

    // Steady state: branch-free, 3 chunks (6 async ops) always in flight.
    int c = 0;
    for (; c + (BUFS - 1) < nchunk; ++c) {
        issue(c + (BUFS - 1));
        wait_async_6();          // in-order completion => chunk c is in LDS
        compute(c);
    }
    // Epilogue: drain the last (up to) 3 chunks.
    for (; c < nchunk; ++c) {
        const int ahead = nchunk - 1 - c;
        if      (ahead >= 2) wait_async_4();
        else if (ahead == 1) wait_async_2();
        else                 wait_async_0();
        compute(c);
    }

    // wave32 cross-lane (min d, tie -> min g) reduction: matches jnp.argmin's
    // first-minimum semantics (per-lane scan already kept the earliest g).
    for (int off = 16; off > 0; off >>= 1) {
        float od = __shfl_xor(bestd, off, WAVE);
        int   og = __shfl_xor(bestg, off, WAVE);
        if (od < bestd || (od == bestd && og < bestg)) { bestd = od; bestg = og; }
    }

    if (lane == 0) {
        out[2 * b]     = Ugrid[bestg];
        out[2 * b + 1] = Ugrid[G + bestg];
    }
}

extern "C" void kernel_launch(void* const* d_in, const int* in_sizes, int n_in,
                              void* d_out, int out_size, void* d_ws, size_t ws_size,
                              hipStream_t stream) {
    const float* y     = (const float*)d_in[0];   // [B,2]
    const float* Ysurf = (const float*)d_in[1];   // [B,2,G]
    const float* Ugrid = (const float*)d_in[2];   // [2,G]
    float* out = (float*)d_out;                   // [B,2]

    const int D = 2;
    const int B = in_sizes[0] / D;                // 16384
    const int G = in_sizes[2] / D;                // 2500

    const int grid = (B + WAVES_PER_BLOCK - 1) / WAVES_PER_BLOCK;
    hipLaunchKernelGGL(vq_push_argmin_kernel, dim3(grid), dim3(BLOCK), 0, stream,
                       y, Ysurf, Ugrid, out, B, G);
}